// ST_block_6803228197053
// MI455X (gfx1250) — compile-verified
//
#include <hip/hip_runtime.h>
#include <math.h>

// ---------------- problem constants ----------------
constexpr int B_ = 8, N_ = 325, T_ = 12, C_ = 64, H_ = 4, D_ = 16;
constexpr int C2_ = 128, C4_ = 256;
constexpr int NN_ = N_ * N_;                 // 105625
constexpr int RROWS_ = B_ * N_ * T_;         // 31200 rows of C
constexpr int RC_ = RROWS_ * C_;             // 1996800 floats per (B,N,T,C) tensor

typedef float v2f __attribute__((ext_vector_type(2)));
typedef float v8f __attribute__((ext_vector_type(8)));

#define FLAG_ACC  1
#define FLAG_RELU 2

#define WMMA_F32(a, b, c) \
    __builtin_amdgcn_wmma_f32_16x16x4_f32(false, (a), false, (b), (short)0, (c), false, false)

// =====================================================================
// Generic batched/strided WMMA FP32 GEMM:  C[bz] (+)= A[bz] @ B[bz] (+bias)
// One wave computes a 32x32 tile (2x2 of 16x16 WMMA accumulators).
// Loads are branch-free: OOB rows/cols are CLAMPED (their results are
// never stored); only the K%4 tail needs value masking.
// =====================================================================
__global__ __launch_bounds__(32)
void k_gemm(const float* __restrict__ A, const float* __restrict__ Bm,
            float* __restrict__ Cm, const float* __restrict__ bias,
            int M, int Kd, int Ncols, int lda, int ldb, int ldc,
            long strideA, long strideB, long strideC, int flags)
{
    const int lane = threadIdx.x;
    const int l16  = lane & 15;
    const int half = lane >> 4;
    const int m0 = blockIdx.x * 32;
    const int n0 = blockIdx.y * 32;
    const long bz = blockIdx.z;
    A  += bz * strideA;
    Bm += bz * strideB;
    Cm += bz * strideC;

    // clamped lane rows/cols (branch-free addressing; OOB never stored)
    const int am0 = min(m0 + l16,      M - 1);
    const int am1 = min(m0 + 16 + l16, M - 1);
    const int bn0 = min(n0 + l16,      Ncols - 1);
    const int bn1 = min(n0 + 16 + l16, Ncols - 1);

    const float* a0p = A + (long)am0 * lda + 2 * half;
    const float* a1p = A + (long)am1 * lda + 2 * half;
    const float* b0p = Bm + (long)(2 * half) * ldb + bn0;
    const float* b1p = Bm + (long)(2 * half) * ldb + bn1;
    const long bstep = (long)4 * ldb;

    v8f acc00 = {}, acc01 = {}, acc10 = {}, acc11 = {};
    const int Kd4 = Kd & ~3;
    for (int kk = 0; kk < Kd4; kk += 4) {
        v2f a0, a1, b0, b1;
        a0.x = a0p[0];   a0.y = a0p[1];
        a1.x = a1p[0];   a1.y = a1p[1];
        b0.x = b0p[0];   b0.y = b0p[ldb];
        b1.x = b1p[0];   b1.y = b1p[ldb];
        acc00 = WMMA_F32(a0, b0, acc00);
        acc01 = WMMA_F32(a0, b1, acc01);
        acc10 = WMMA_F32(a1, b0, acc10);
        acc11 = WMMA_F32(a1, b1, acc11);
        a0p += 4; a1p += 4; b0p += bstep; b1p += bstep;
    }
    if (Kd4 < Kd) {                      // K tail (e.g. Kd = 325)
        const int ka = Kd4 + 2 * half;
        const int k0 = min(ka,     Kd - 1);
        const int k1 = min(ka + 1, Kd - 1);
        const float w0 = (ka     < Kd) ? 1.f : 0.f;
        const float w1 = (ka + 1 < Kd) ? 1.f : 0.f;
        v2f a0, a1, b0, b1;
        a0.x = A[(long)am0 * lda + k0] * w0;  a0.y = A[(long)am0 * lda + k1] * w1;
        a1.x = A[(long)am1 * lda + k0] * w0;  a1.y = A[(long)am1 * lda + k1] * w1;
        b0.x = Bm[(long)k0 * ldb + bn0];      b0.y = Bm[(long)k1 * ldb + bn0];
        b1.x = Bm[(long)k0 * ldb + bn1];      b1.y = Bm[(long)k1 * ldb + bn1];
        acc00 = WMMA_F32(a0, b0, acc00);
        acc01 = WMMA_F32(a0, b1, acc01);
        acc10 = WMMA_F32(a1, b0, acc10);
        acc11 = WMMA_F32(a1, b1, acc11);
    }

    // C/D layout: VGPR j -> row mb + j + 8*half, col nb + l16
    v8f accs[4] = { acc00, acc01, acc10, acc11 };
    #pragma unroll
    for (int tile = 0; tile < 4; ++tile) {
        const int mb = m0 + (tile >> 1) * 16;
        const int nb = n0 + (tile & 1) * 16;
        const int n = nb + l16;
        #pragma unroll
        for (int j = 0; j < 8; ++j) {
            const int m = mb + j + 8 * half;
            if (m < M && n < Ncols) {
                float v = accs[tile][j];
                if (bias)              v += bias[n];
                if (flags & FLAG_ACC)  v += Cm[(long)m * ldc + n];
                if (flags & FLAG_RELU) v = fmaxf(v, 0.f);
                Cm[(long)m * ldc + n] = v;
            }
        }
    }
}

// =====================================================================
// Flash attention over nodes for one (b,h,t, 16-row query tile).
// D = 16 exactly -> the running context is one 16x16 WMMA accumulator.
// Tensors stored (B,N,T,C) with head offset h*D.
// =====================================================================
__device__ __forceinline__ float grp_max16(float v) {
    #pragma unroll
    for (int m = 1; m < 16; m <<= 1) v = fmaxf(v, __shfl_xor(v, m, 16));
    return v;
}
__device__ __forceinline__ float grp_sum16(float v) {
    #pragma unroll
    for (int m = 1; m < 16; m <<= 1) v += __shfl_xor(v, m, 16);
    return v;
}

__global__ __launch_bounds__(32)
void k_flash_attn(const float* __restrict__ Q, const float* __restrict__ Kmat,
                  const float* __restrict__ V, float* __restrict__ Out)
{
    __shared__ float lds[16 * 16];

    const int lane = threadIdx.x;
    const int l16  = lane & 15;
    const int half = lane >> 4;

    const int n0  = blockIdx.x * 16;        // query-node tile
    const int bht = blockIdx.y;             // b*H*T + h*T + t
    const int t = bht % T_;
    const int h = (bht / T_) % H_;
    const int b = bht / (T_ * H_);

    const long rowstride = (long)T_ * C_;   // node stride in (B,N,T,C)
    const long base = ((long)(b * N_) * T_ + t) * C_ + h * D_;
    const float scale = 0.25f;              // 1/sqrt(D)

    // Q tile as A operand (row = query node, k = d); clamped, vector loads
    const int qrow = min(n0 + l16, N_ - 1);
    const float* qp = Q + base + (long)qrow * rowstride;
    v2f qa[4];
    #pragma unroll
    for (int s = 0; s < 4; ++s)
        qa[s] = *(const v2f*)(qp + s * 4 + 2 * half);

    v8f o = {};
    float rm[8], rl[8];
    #pragma unroll
    for (int j = 0; j < 8; ++j) { rm[j] = -3.0e38f; rl[j] = 0.f; }

    const int mtiles = (N_ + 15) / 16;      // 21
    for (int mt = 0; mt < mtiles; ++mt) {
        const int m0 = mt * 16;

        // ---- S = Q @ K^T : B operand (k=d, n=key node), clamped rows ----
        const int krow = min(m0 + l16, N_ - 1);
        const float* kp = Kmat + base + (long)krow * rowstride;
        v8f s = {};
        #pragma unroll
        for (int st = 0; st < 4; ++st) {
            const v2f kb = *(const v2f*)(kp + st * 4 + 2 * half);
            s = WMMA_F32(qa[st], kb, s);
        }

        // ---- online softmax (rows of S spread across 16-lane groups) ----
        const bool col_ok = (m0 + l16) < N_;
        float p[8];
        #pragma unroll
        for (int j = 0; j < 8; ++j) {
            float v = col_ok ? s[j] * scale : -1.0e30f;
            const float tmax = grp_max16(v);
            const float nm = fmaxf(rm[j], tmax);
            const float corr = __expf(rm[j] - nm);
            p[j] = __expf(v - nm);
            const float ps = grp_sum16(p[j]);
            rl[j] = rl[j] * corr + ps;
            rm[j] = nm;
            o[j] *= corr;
        }

        // ---- stage P through LDS: C-layout -> A-layout ----
        #pragma unroll
        for (int j = 0; j < 8; ++j)
            lds[(j + 8 * half) * 16 + l16] = p[j];
        __syncthreads();

        // ---- O += P @ V : A from LDS, B = V tile (k=key node, n=d) ----
        #pragma unroll
        for (int st = 0; st < 4; ++st) {
            const int kk = st * 4 + 2 * half;
            v2f a, vb;
            a.x = lds[l16 * 16 + kk];
            a.y = lds[l16 * 16 + kk + 1];
            const int vr0 = min(m0 + kk,     N_ - 1);   // clamped; P cols are 0 there
            const int vr1 = min(m0 + kk + 1, N_ - 1);
            vb.x = V[base + (long)vr0 * rowstride + l16];
            vb.y = V[base + (long)vr1 * rowstride + l16];
            o = WMMA_F32(a, vb, o);
        }
        __syncthreads();
    }

    // normalize and store ctx (B,N,T,C) layout
    #pragma unroll
    for (int j = 0; j < 8; ++j) {
        const int n = n0 + j + 8 * half;
        if (n < N_)
            Out[base + (long)n * rowstride + l16] = o[j] / rl[j];
    }
}

// =====================================================================
// Elementwise / reduction helper kernels
// =====================================================================
__device__ __forceinline__ float pe_val(int pos, int c) {
    const float e = (float)(2 * (c / 2)) / (float)C_;
    const float ang = (float)pos * __powf(10000.f, -e);
    return (c & 1) ? __cosf(ang) : __sinf(ang);
}

__global__ void k_pe_add(const float* __restrict__ fin, const float* __restrict__ sin_,
                         float* __restrict__ fq, float* __restrict__ sq, int total)
{
    int i = blockIdx.x * blockDim.x + threadIdx.x;
    if (i >= total) return;
    const int c = i % C_;
    const int t = (i / C_) % T_;
    const int n = (i / (C_ * T_)) % N_;
    const float p = pe_val(n, c) + pe_val(t, c);
    fq[i] = fin[i] + p;
    sq[i] = sin_[i] + p;
}

__global__ __launch_bounds__(256)
void k_adj_stats(const float* __restrict__ adj, float* __restrict__ stats)
{
    __shared__ float ss[256], sq2[256];
    float s = 0.f, q = 0.f;
    for (int i = threadIdx.x; i < NN_; i += 256) { float v = adj[i]; s += v; q += v * v; }
    ss[threadIdx.x] = s; sq2[threadIdx.x] = q;
    __syncthreads();
    for (int off = 128; off > 0; off >>= 1) {
        if (threadIdx.x < off) { ss[threadIdx.x] += ss[threadIdx.x + off];
                                 sq2[threadIdx.x] += sq2[threadIdx.x + off]; }
        __syncthreads();
    }
    if (threadIdx.x == 0) {
        const float m = ss[0] / (float)NN_;
        const float var = sq2[0] / (float)NN_ - m * m;
        stats[0] = m;
        stats[1] = rsqrtf(var + 1e-5f);
    }
}

__global__ void k_cheb_init(const float* __restrict__ adj, const float* __restrict__ stats,
                            float* __restrict__ cheb)
{
    int i = blockIdx.x * blockDim.x + threadIdx.x;
    if (i >= NN_) return;
    const int r = i / N_, c = i % N_;
    cheb[i]        = (r == c) ? 1.f : 0.f;            // T0 = I
    cheb[NN_ + i]  = (adj[i] - stats[0]) * stats[1];  // T1 = adjn
}

__global__ void k_cheb2_fix(float* __restrict__ cheb2)
{
    int i = blockIdx.x * blockDim.x + threadIdx.x;
    if (i >= NN_) return;
    const int r = i / N_, c = i % N_;
    cheb2[i] = 2.f * cheb2[i] - ((r == c) ? 1.f : 0.f);   // T2 = 2*A@T1 - T0
}

__global__ void k_qfs(const float* __restrict__ Qs, const float* __restrict__ Kj,
                      const float* __restrict__ Vfp, float* __restrict__ out, int total)
{
    int i = blockIdx.x * blockDim.x + threadIdx.x;
    if (i >= total) return;
    const int n = (i / (T_ * C_)) % N_;
    const float q = Qs[i];
    out[i] = Kj[n] * (q - q * q / (Vfp[n] + 1e-5f));
}

__global__ void k_gate_combine(const float* __restrict__ g, const float* __restrict__ a,
                               const float* __restrict__ b2, float* __restrict__ out, int total)
{
    int i = blockIdx.x * blockDim.x + threadIdx.x;
    if (i >= total) return;
    const float s = 1.f / (1.f + __expf(-g[i]));
    out[i] = s * a[i] + (1.f - s) * b2[i];
}

// circular 5-tap mean over T (collapses the WIN shift loop)
__global__ void k_window(const float* __restrict__ src, float* __restrict__ dst, int total)
{
    int i = blockIdx.x * blockDim.x + threadIdx.x;
    if (i >= total) return;
    const int c = i % C_;
    const int t = (i / C_) % T_;
    const long bn = (long)(i / (T_ * C_)) * T_ * C_;
    float s = 0.f;
    #pragma unroll
    for (int d = -2; d <= 2; ++d) {
        int tt = t + d;
        if (tt < 0) tt += T_;
        if (tt >= T_) tt -= T_;
        s += src[bn + (long)tt * C_ + c];
    }
    dst[i] = s * 0.2f;
}

// out = LN(x + add) * g + b, per row of C=64 (one wave per row)
__global__ __launch_bounds__(32)
void k_layernorm(const float* __restrict__ x, const float* __restrict__ add,
                 const float* __restrict__ g, const float* __restrict__ bgain,
                 float* __restrict__ out)
{
    const long row = blockIdx.x;
    const int lane = threadIdx.x;
    const float v0 = x[row * C_ + lane]      + add[row * C_ + lane];
    const float v1 = x[row * C_ + lane + 32] + add[row * C_ + lane + 32];
    float s = v0 + v1, q = v0 * v0 + v1 * v1;
    #pragma unroll
    for (int m = 1; m < 32; m <<= 1) { s += __shfl_xor(s, m, 32); q += __shfl_xor(q, m, 32); }
    const float mean = s * (1.f / 64.f);
    const float var  = q * (1.f / 64.f) - mean * mean;
    const float inv  = rsqrtf(var + 1e-5f);
    out[row * C_ + lane]      = (v0 - mean) * inv * g[lane]      + bgain[lane];
    out[row * C_ + lane + 32] = (v1 - mean) * inv * g[lane + 32] + bgain[lane + 32];
}

__global__ void k_final(const float* __restrict__ of, const float* __restrict__ os,
                        const float* __restrict__ w, const float* __restrict__ bb,
                        float* __restrict__ out, int total)
{
    int i = blockIdx.x * blockDim.x + threadIdx.x;
    if (i >= total) return;
    out[i] = of[i] * w[0] + os[i] * w[1] + bb[0];
}

// =====================================================================
// Host orchestration
// =====================================================================
extern "C" void kernel_launch(void* const* d_in, const int* in_sizes, int n_in,
                              void* d_out, int out_size, void* d_ws, size_t ws_size,
                              hipStream_t stream)
{
    const float* flow_query  = (const float*)d_in[0];
    const float* flow_key    = (const float*)d_in[1];
    const float* flow_value  = (const float*)d_in[2];
    const float* speed_query = (const float*)d_in[3];
    const float* speed_key   = (const float*)d_in[4];
    const float* speed_value = (const float*)d_in[5];
    const float* adj         = (const float*)d_in[6];
    const float* WfQ = (const float*)d_in[7];
    const float* WfK = (const float*)d_in[8];
    const float* WfV = (const float*)d_in[9];
    const float* WsQ = (const float*)d_in[10];
    const float* WsK = (const float*)d_in[11];
    const float* WsV = (const float*)d_in[12];
    const float* Kj   = (const float*)d_in[13];
    const float* Vf_p = (const float*)d_in[14];
    const float* fc_out_W = (const float*)d_in[15];
    const float* fc_out_b = (const float*)d_in[16];
    const float* flow_gate_W = (const float*)d_in[17];
    const float* flow_gate_b = (const float*)d_in[18];
    // d_in[19], d_in[20]: speed gate — mathematically dead in the reference
    const float* norm1_g = (const float*)d_in[21];
    const float* norm1_b = (const float*)d_in[22];
    const float* norm2_g = (const float*)d_in[23];
    const float* norm2_b = (const float*)d_in[24];
    const float* ff_W1 = (const float*)d_in[25];
    const float* ff_b1 = (const float*)d_in[26];
    const float* ff_W2 = (const float*)d_in[27];
    const float* ff_b2 = (const float*)d_in[28];
    const float* fs_W = (const float*)d_in[29];
    const float* fs_b = (const float*)d_in[30];
    const float* fg_W = (const float*)d_in[31];
    const float* fg_b = (const float*)d_in[32];
    const float* cheb_W1 = (const float*)d_in[33];   // (3,64,128)
    const float* cheb_W2 = (const float*)d_in[34];   // (3,128,64)
    const float* lin2_W  = (const float*)d_in[35];
    const float* lin2_b  = (const float*)d_in[36];

    float* ws = (float*)d_ws;
    auto slot = [&](int i) { return ws + (long)i * RC_; };
    float* fq = slot(0);  float* sq = slot(1);
    float* fXG = slot(2); float* sXG = slot(3);
    float* Qf = slot(4);  float* Kf = slot(5);  float* Vf = slot(6);
    float* Qs = slot(7);  float* Ks = slot(8);  float* Vs = slot(9);
    float* Qfs = slot(10);
    float* o_ff = slot(11); float* o_fs = slot(12); float* o_ss = slot(13);
    float* res_f = slot(14); float* res_s = slot(15);
    float* out_flow = slot(16); float* out_speed = slot(17);
    float* y = slot(18);              // RC
    float* hbuf = slot(19);           // 2*RC
    float* zbuf = slot(21);           // 2*RC
    float* cheb  = ws + (long)23 * RC_;   // 3*NN
    float* stats = cheb + 3 * NN_;        // 2 floats

    auto gemm = [&](const float* A, const float* Bm, float* Cm, const float* bias,
                    int M, int Kd, int Nc, int lda, int ldb, int ldc,
                    int batch, long sA, long sB, long sC, int flags) {
        dim3 g((M + 31) / 32, (Nc + 31) / 32, batch);
        k_gemm<<<g, 32, 0, stream>>>(A, Bm, Cm, bias, M, Kd, Nc, lda, ldb, ldc,
                                     sA, sB, sC, flags);
    };
    const int EB = 256;
    const int egrid = (RC_ + EB - 1) / EB;

    // 1) positional encodings
    k_pe_add<<<egrid, EB, 0, stream>>>(flow_query, speed_query, fq, sq, RC_);

    // 2) Chebyshev basis
    k_adj_stats<<<1, 256, 0, stream>>>(adj, stats);
    k_cheb_init<<<(NN_ + EB - 1) / EB, EB, 0, stream>>>(adj, stats, cheb);
    gemm(cheb + NN_, cheb + NN_, cheb + 2 * NN_, nullptr,
         N_, N_, N_, N_, N_, N_, 1, 0, 0, 0, 0);
    k_cheb2_fix<<<(NN_ + EB - 1) / EB, EB, 0, stream>>>(cheb + 2 * NN_);

    // 3) GCN branches: xg = sum_k cheb_k @ relu(sum_k cheb_k @ x @ W1_k) @ W2_k
    const long sbx = (long)N_ * T_ * C_;      // per-batch stride of x (325x768)
    const long sbh = (long)N_ * T_ * C2_;     // per-batch stride of h (325x1536)
    const float* gx[2] = { fq, sq };
    float* gout[2] = { fXG, sXG };
    for (int br = 0; br < 2; ++br) {
        for (int k = 0; k < 3; ++k) {
            gemm(cheb + (long)k * NN_, gx[br], y, nullptr,
                 N_, N_, T_ * C_, N_, T_ * C_, T_ * C_, B_, 0, sbx, sbx, 0);
            int fl = (k ? FLAG_ACC : 0) | (k == 2 ? FLAG_RELU : 0);
            gemm(y, cheb_W1 + (long)k * C_ * C2_, hbuf, nullptr,
                 RROWS_, C_, C2_, C_, C2_, C2_, 1, 0, 0, 0, fl);
        }
        for (int k = 0; k < 3; ++k) {
            gemm(cheb + (long)k * NN_, hbuf, zbuf, nullptr,
                 N_, N_, T_ * C2_, N_, T_ * C2_, T_ * C2_, B_, 0, sbh, sbh, 0);
            gemm(zbuf, cheb_W2 + (long)k * C2_ * C_, gout[br], nullptr,
                 RROWS_, C2_, C_, C2_, C_, C_, 1, 0, 0, 0, k ? FLAG_ACC : 0);
        }
    }

    // 4) projections (kept in (B,N,T,C); heads addressed by stride in attention)
    gemm(fq,          WfQ, Qf, nullptr, RROWS_, C_, C_, C_, C_, C_, 1, 0,0,0, 0);
    gemm(flow_key,    WfK, Kf, nullptr, RROWS_, C_, C_, C_, C_, C_, 1, 0,0,0, 0);
    gemm(flow_value,  WfV, Vf, nullptr, RROWS_, C_, C_, C_, C_, C_, 1, 0,0,0, 0);
    gemm(sq,          WsQ, Qs, nullptr, RROWS_, C_, C_, C_, C_, C_, 1, 0,0,0, 0);
    gemm(speed_key,   WsK, Ks, nullptr, RROWS_, C_, C_, C_, C_, C_, 1, 0,0,0, 0);
    gemm(speed_value, WsV, Vs, nullptr, RROWS_, C_, C_, C_, C_, C_, 1, 0,0,0, 0);
    k_qfs<<<egrid, EB, 0, stream>>>(Qs, Kj, Vf_p, Qfs, RC_);

    // 5) three attentions (base time only; WIN loop == circular window)
    float* ctx  = slot(18);
    float* gbuf = slot(19);
    float* comb = slot(20);
    dim3 agrid((N_ + 15) / 16, B_ * H_ * T_);
    k_flash_attn<<<agrid, 32, 0, stream>>>(Qf, Kf, Vf, ctx);
    gemm(ctx, fc_out_W, o_ff, fc_out_b, RROWS_, C_, C_, C_, C_, C_, 1, 0,0,0, 0);
    k_flash_attn<<<agrid, 32, 0, stream>>>(Kf, Qfs, Vf, ctx);
    gemm(ctx, fc_out_W, o_fs, fc_out_b, RROWS_, C_, C_, C_, C_, C_, 1, 0,0,0, 0);
    k_flash_attn<<<agrid, 32, 0, stream>>>(Qs, Ks, Vs, ctx);
    gemm(ctx, fc_out_W, o_ss, fc_out_b, RROWS_, C_, C_, C_, C_, C_, 1, 0,0,0, 0);

    // 6) flow gate + circular 5-tap window (speed gate cancels: res_speed = mean o_ss)
    gemm(o_ff, flow_gate_W,            gbuf, nullptr,     RROWS_, C_, C_, C_, C_, C_, 1, 0,0,0, 0);
    gemm(o_fs, flow_gate_W + C_ * C_,  gbuf, flow_gate_b, RROWS_, C_, C_, C_, C_, C_, 1, 0,0,0, FLAG_ACC);
    k_gate_combine<<<egrid, EB, 0, stream>>>(gbuf, o_ff, o_fs, comb, RC_);
    k_window<<<egrid, EB, 0, stream>>>(comb, res_f, RC_);
    k_window<<<egrid, EB, 0, stream>>>(o_ss, res_s, RC_);

    // 7) output branches (reuse freed projection slots)
    float* x1   = slot(4);
    float* ff1  = slot(5);   // 4 contiguous RC slots (R x 256)
    float* ff2  = slot(9);
    float* u    = slot(10);
    float* gacc = slot(18);
    const float* resv[2] = { res_f, res_s };
    const float* qpe[2]  = { fq, sq };
    const float* xg[2]   = { fXG, sXG };
    float* bout[2] = { out_flow, out_speed };
    for (int br = 0; br < 2; ++br) {
        k_layernorm<<<RROWS_, 32, 0, stream>>>(resv[br], qpe[br], norm1_g, norm1_b, x1);
        gemm(x1,  ff_W1, ff1, ff_b1, RROWS_, C_,  C4_, C_,  C4_, C4_, 1, 0,0,0, FLAG_RELU);
        gemm(ff1, ff_W2, ff2, ff_b2, RROWS_, C4_, C_,  C4_, C_,  C_,  1, 0,0,0, 0);
        k_layernorm<<<RROWS_, 32, 0, stream>>>(ff2, x1, norm2_g, norm2_b, u);
        gemm(u,      fs_W, gacc, fs_b, RROWS_, C_, C_, C_, C_, C_, 1, 0,0,0, 0);
        gemm(xg[br], fg_W, gacc, fg_b, RROWS_, C_, C_, C_, C_, C_, 1, 0,0,0, FLAG_ACC);
        k_gate_combine<<<egrid, EB, 0, stream>>>(gacc, u, xg[br], bout[br], RC_);
    }

    // 8) final linear combine
    k_final<<<egrid, EB, 0, stream>>>(out_flow, out_speed, lin2_W, lin2_b,
                                      (float*)d_out, RC_);
}